// DynamicGraphEncoder_35931696398655
// MI455X (gfx1250) — compile-verified
//
#include <hip/hip_runtime.h>
#include <math.h>

#define KSNAP  8
#define NNODES 10000
#define NEDGES 160000
#define EPRIME (NEDGES + NNODES)
#define HHEADS 8
#define CHID   32
#define DDIM   256
#define LH     256

typedef float v2f __attribute__((ext_vector_type(2)));
typedef float v8f __attribute__((ext_vector_type(8)));

__device__ __forceinline__ int f2ord(float f) {
  int i = __float_as_int(f);
  return i >= 0 ? i : (i ^ 0x7FFFFFFF);
}
__device__ __forceinline__ float ord2f(int i) {
  return __int_as_float(i >= 0 ? i : (i ^ 0x7FFFFFFF));
}
__device__ __forceinline__ float sigmoidf_(float x) { return 1.0f / (1.0f + __expf(-x)); }

// ---------------- Layer-1 features (D1==1: rank-1 product) ----------------
__global__ void k_l1_feat(const float* __restrict__ x, const float* __restrict__ W1,
                          const float* __restrict__ a_src, const float* __restrict__ a_dst,
                          float* __restrict__ h, float* __restrict__ ssrc, float* __restrict__ sdst) {
  int idx = blockIdx.x * blockDim.x + threadIdx.x;   // n*256 + hc
  if (idx >= NNODES * DDIM) return;
  int n = idx >> 8, hc = idx & 255;
  float xv = x[n];
  h[idx] = xv * W1[hc];
  if (hc < HHEADS) {
    float cs = 0.f, cd = 0.f;
    for (int c = 0; c < CHID; ++c) {
      float w = W1[hc * CHID + c];
      cs += w * a_src[hc * CHID + c];
      cd += w * a_dst[hc * CHID + c];
    }
    ssrc[n * HHEADS + hc] = xv * cs;
    sdst[n * HHEADS + hc] = xv * cd;
  }
}

// ---------------- Per-layer init: acc=0, emax=-inf(ordered), denom=0 ----------------
__global__ void k_init(float* __restrict__ acc, int* __restrict__ emax, float* __restrict__ denom) {
  int idx = blockIdx.x * blockDim.x + threadIdx.x;
  if (idx < NNODES * DDIM) acc[idx] = 0.f;
  if (idx < NNODES * HHEADS) {
    emax[idx] = f2ord(-INFINITY);
    denom[idx] = 0.f;
  }
}

// ---------------- Edge pass 1: e = leaky_relu(s_src[src]+s_dst[dst]); segment max ----------------
__global__ void k_edge_max(const int* __restrict__ ei, const float* __restrict__ ssrc,
                           const float* __restrict__ sdst, float* __restrict__ ebuf,
                           int* __restrict__ emax) {
  int idx = blockIdx.x * blockDim.x + threadIdx.x;   // e*8 + head
  if (idx >= EPRIME * HHEADS) return;
  int e = idx >> 3, hh = idx & 7;
  int s, d;
  if (e < NEDGES) { s = ei[e]; d = ei[NEDGES + e]; } else { s = d = e - NEDGES; }
  float v = ssrc[s * HHEADS + hh] + sdst[d * HHEADS + hh];
  v = v > 0.f ? v : 0.2f * v;            // leaky_relu, PyG slope
  ebuf[idx] = v;
  atomicMax(&emax[d * HHEADS + hh], f2ord(v));
}

// ---------------- Edge pass 2: ex = exp(e - max); segment sum ----------------
__global__ void k_edge_sum(const int* __restrict__ ei, const int* __restrict__ emax,
                           float* __restrict__ ebuf, float* __restrict__ denom) {
  int idx = blockIdx.x * blockDim.x + threadIdx.x;
  if (idx >= EPRIME * HHEADS) return;
  int e = idx >> 3, hh = idx & 7;
  int d;
  if (e < NEDGES) { d = ei[NEDGES + e]; } else { d = e - NEDGES; }
  float ex = __expf(ebuf[idx] - ord2f(emax[d * HHEADS + hh]));
  ebuf[idx] = ex;
  atomicAdd(&denom[d * HHEADS + hh], ex);
}

// ---------------- Edge pass 3: acc[dst] += alpha * h[src] (block per edge, thread per channel) ----------------
__global__ void k_edge_agg(const int* __restrict__ ei, const float* __restrict__ h,
                           const float* __restrict__ ebuf, const float* __restrict__ denom,
                           float* __restrict__ acc) {
  int e = blockIdx.x;
  int hc = threadIdx.x;                  // 0..255
  int s, d;
  if (e < NEDGES) { s = ei[e]; d = ei[NEDGES + e]; } else { s = d = e - NEDGES; }
  int hh = hc >> 5;
  float alpha = ebuf[e * HHEADS + hh] / (denom[d * HHEADS + hh] + 1e-16f);
  atomicAdd(&acc[d * DDIM + hc], alpha * h[s * DDIM + hc]);
}

// ---------------- bias + relu ----------------
__global__ void k_bias_relu(const float* __restrict__ acc, const float* __restrict__ b,
                            float* __restrict__ out) {
  int idx = blockIdx.x * blockDim.x + threadIdx.x;
  if (idx >= NNODES * DDIM) return;
  float v = acc[idx] + b[idx & 255];
  out[idx] = v > 0.f ? v : 0.f;
}

// ---------------- Layer-2 GEMM: C[M,256] = A[M,256] x B[256,256] via V_WMMA_F32_16X16X4_F32 ----
// One 16x16 C tile per wave32; K-loop in steps of 4 (exact f32 accumulation).
// A layout (16x4, ISA 7.12.2): lanes 0-15 hold row M=lane, K={k0,k0+1}; lanes 16-31 K={k0+2,k0+3}.
// C/D layout: VGPR v -> M = v + 8*lane[4], N = lane[3:0].
__global__ void k_gemm_wmma(const float* __restrict__ A, const float* __restrict__ B,
                            float* __restrict__ C, int M) {
  int wave = threadIdx.x >> 5;             // 8 waves per block
  int lane = threadIdx.x & 31;
  int tilesN = DDIM / 16;                  // 16
  int tileId = blockIdx.x * 8 + wave;
  int tm = tileId / tilesN;
  int tn = tileId % tilesN;
  if (tm * 16 >= M) return;                // wave-uniform guard (EXEC stays all-1s)

  int l15 = lane & 15;
  int khalf = (lane >> 4) * 2;             // 0 or 2
  int arow = tm * 16 + l15;
  int bcol = tn * 16 + l15;

  v8f c = {};
  for (int k0 = 0; k0 < DDIM; k0 += 4) {
    v2f a, b;
    a.x = A[arow * DDIM + k0 + khalf];
    a.y = A[arow * DDIM + k0 + khalf + 1];
    b.x = B[(k0 + khalf) * DDIM + bcol];
    b.y = B[(k0 + khalf + 1) * DDIM + bcol];
    c = __builtin_amdgcn_wmma_f32_16x16x4_f32(false, a, false, b, (short)0, c, false, false);
  }
  int crow = tm * 16 + (lane >> 4) * 8;
  int ccol = tn * 16 + l15;
  for (int v = 0; v < 8; ++v)
    C[(crow + v) * DDIM + ccol] = c[v];
}

// ---------------- attention scores for layer 2 ----------------
__global__ void k_attn_scores(const float* __restrict__ h, const float* __restrict__ a_src,
                              const float* __restrict__ a_dst, float* __restrict__ ssrc,
                              float* __restrict__ sdst) {
  int idx = blockIdx.x * blockDim.x + threadIdx.x;   // n*8 + head
  if (idx >= NNODES * HHEADS) return;
  int n = idx >> 3, hh = idx & 7;
  const float* hp = h + n * DDIM + hh * CHID;
  float cs = 0.f, cd = 0.f;
  for (int c = 0; c < CHID; ++c) {
    cs += hp[c] * a_src[hh * CHID + c];
    cd += hp[c] * a_dst[hh * CHID + c];
  }
  ssrc[idx] = cs;
  sdst[idx] = cd;
}

// ---------------- zero embs ----------------
__global__ void k_zero(float* __restrict__ p, int n) {
  int idx = blockIdx.x * blockDim.x + threadIdx.x;
  if (idx < n) p[idx] = 0.f;
}

// ---------------- mean pool of relu(acc + b2) into emb[k] ----------------
__global__ void k_pool(const float* __restrict__ acc, const float* __restrict__ b,
                       float* __restrict__ emb) {
  int t = threadIdx.x;                     // channel
  int r0 = blockIdx.x * 64;
  int rend = r0 + 64; if (rend > NNODES) rend = NNODES;
  float s = 0.f;
  float bv = b[t];
  for (int r = r0; r < rend; ++r) {
    float v = acc[r * DDIM + t] + bv;
    s += v > 0.f ? v : 0.f;
  }
  atomicAdd(&emb[t], s * (1.0f / (float)NNODES));
}

// ---------------- bidirectional LSTM over K=8 embeddings (single block, 256 threads) ----------------
__global__ void k_lstm(const float* __restrict__ embs,
                       const float* __restrict__ Wih_f, const float* __restrict__ Whh_f,
                       const float* __restrict__ bih_f, const float* __restrict__ bhh_f,
                       const float* __restrict__ Wih_b, const float* __restrict__ Whh_b,
                       const float* __restrict__ bih_b, const float* __restrict__ bhh_b,
                       float* __restrict__ out) {
  __shared__ float sh[LH], sc[LH], sx[DDIM];
  int t = threadIdx.x;
  for (int dir = 0; dir < 2; ++dir) {
    const float* Wih = dir ? Wih_b : Wih_f;
    const float* Whh = dir ? Whh_b : Whh_f;
    const float* bih = dir ? bih_b : bih_f;
    const float* bhh = dir ? bhh_b : bhh_f;
    sh[t] = 0.f; sc[t] = 0.f;
    __syncthreads();
    for (int s = 0; s < KSNAP; ++s) {
      int k = dir ? (KSNAP - 1 - s) : s;
      sx[t] = embs[k * DDIM + t];
      __syncthreads();
      float g[4];
      for (int q = 0; q < 4; ++q) {       // rows t, t+256, t+512, t+768 -> i,f,g,o of unit t
        int row = q * LH + t;
        float acc = bih[row] + bhh[row];
        const float* wi = Wih + (size_t)row * DDIM;
        const float* wh = Whh + (size_t)row * LH;
        for (int d = 0; d < DDIM; ++d) acc += sx[d] * wi[d];
        for (int d = 0; d < LH; ++d)   acc += sh[d] * wh[d];
        g[q] = acc;
      }
      __syncthreads();                     // all reads of sh done before update
      float ig = sigmoidf_(g[0]);
      float fg = sigmoidf_(g[1]);
      float gg = tanhf(g[2]);
      float og = sigmoidf_(g[3]);
      float c = fg * sc[t] + ig * gg;
      sc[t] = c;
      sh[t] = og * tanhf(c);
      __syncthreads();
    }
    out[dir * LH + t] = sh[t];
    __syncthreads();
  }
}

extern "C" void kernel_launch(void* const* d_in, const int* in_sizes, int n_in,
                              void* d_out, int out_size, void* d_ws, size_t ws_size,
                              hipStream_t stream) {
  (void)in_sizes; (void)n_in; (void)out_size; (void)ws_size;
  const float* xs      = (const float*)d_in[0];
  const int*   ei_all  = (const int*)  d_in[1];
  const float* W1      = (const float*)d_in[2];
  const float* a_src1  = (const float*)d_in[3];
  const float* a_dst1  = (const float*)d_in[4];
  const float* b1      = (const float*)d_in[5];
  const float* W2      = (const float*)d_in[6];
  const float* a_src2  = (const float*)d_in[7];
  const float* a_dst2  = (const float*)d_in[8];
  const float* b2      = (const float*)d_in[9];
  const float* Wih_f   = (const float*)d_in[10];
  const float* Whh_f   = (const float*)d_in[11];
  const float* bih_f   = (const float*)d_in[12];
  const float* bhh_f   = (const float*)d_in[13];
  const float* Wih_b   = (const float*)d_in[14];
  const float* Whh_b   = (const float*)d_in[15];
  const float* bih_b   = (const float*)d_in[16];
  const float* bhh_b   = (const float*)d_in[17];
  float* out = (float*)d_out;

  // Workspace layout (f32 words unless noted)
  char* ws = (char*)d_ws;
  size_t off = 0;
  auto take = [&](size_t bytes) { char* p = ws + off; off += (bytes + 255) & ~size_t(255); return p; };
  float* hbuf  = (float*)take((size_t)NNODES * DDIM * 4);   // GAT features h
  float* hbuf2 = (float*)take((size_t)NNODES * DDIM * 4);   // relu(acc+b1) GEMM input
  float* acc   = (float*)take((size_t)NNODES * DDIM * 4);   // scatter accumulator
  float* ssrc  = (float*)take((size_t)NNODES * HHEADS * 4);
  float* sdst  = (float*)take((size_t)NNODES * HHEADS * 4);
  int*   emax  = (int*)  take((size_t)NNODES * HHEADS * 4);
  float* denom = (float*)take((size_t)NNODES * HHEADS * 4);
  float* ebuf  = (float*)take((size_t)EPRIME * HHEADS * 4);
  float* embs  = (float*)take((size_t)KSNAP * DDIM * 4);

  const int TB = 256;
  const int gND   = (NNODES * DDIM + TB - 1) / TB;           // node*channel grid
  const int gEH   = (EPRIME * HHEADS + TB - 1) / TB;         // edge*head grid
  const int gNH   = (NNODES * HHEADS + TB - 1) / TB;
  const int gGemm = (NNODES / 16) * (DDIM / 16) / 8;         // 8 waves/block, 1 tile/wave
  const int gPool = (NNODES + 63) / 64;

  k_zero<<<(KSNAP * DDIM + TB - 1) / TB, TB, 0, stream>>>(embs, KSNAP * DDIM);

  for (int k = 0; k < KSNAP; ++k) {
    const float* x  = xs + (size_t)k * NNODES;
    const int*   ei = ei_all + (size_t)k * 2 * NEDGES;

    // ---- GAT layer 1 ----
    k_l1_feat<<<gND, TB, 0, stream>>>(x, W1, a_src1, a_dst1, hbuf, ssrc, sdst);
    k_init<<<gND, TB, 0, stream>>>(acc, emax, denom);
    k_edge_max<<<gEH, TB, 0, stream>>>(ei, ssrc, sdst, ebuf, emax);
    k_edge_sum<<<gEH, TB, 0, stream>>>(ei, emax, ebuf, denom);
    k_edge_agg<<<EPRIME, TB, 0, stream>>>(ei, hbuf, ebuf, denom, acc);
    k_bias_relu<<<gND, TB, 0, stream>>>(acc, b1, hbuf2);

    // ---- GAT layer 2 ----
    k_gemm_wmma<<<gGemm, TB, 0, stream>>>(hbuf2, W2, hbuf, NNODES);
    k_attn_scores<<<gNH, TB, 0, stream>>>(hbuf, a_src2, a_dst2, ssrc, sdst);
    k_init<<<gND, TB, 0, stream>>>(acc, emax, denom);
    k_edge_max<<<gEH, TB, 0, stream>>>(ei, ssrc, sdst, ebuf, emax);
    k_edge_sum<<<gEH, TB, 0, stream>>>(ei, emax, ebuf, denom);
    k_edge_agg<<<EPRIME, TB, 0, stream>>>(ei, hbuf, ebuf, denom, acc);

    // ---- mean pool of relu(acc + b2) ----
    k_pool<<<gPool, TB, 0, stream>>>(acc, b2, embs + (size_t)k * DDIM);
  }

  // ---- bidirectional LSTM over the K snapshot embeddings ----
  k_lstm<<<1, LH, 0, stream>>>(embs, Wih_f, Whh_f, bih_f, bhh_f,
                               Wih_b, Whh_b, bih_b, bhh_b, out);
}